// GATv2_63900523430530
// MI455X (gfx1250) — compile-verified
//
#include <hip/hip_runtime.h>
#include <math.h>

#define N_NODES 50000
#define N_EDGES 800000
#define F_IN    128
#define H1      8
#define C1      8
#define HC1     64      // H1*C1
#define NC      10
#define NEG_SLOPE 0.2f

typedef __attribute__((ext_vector_type(2))) float v2f;
typedef __attribute__((ext_vector_type(8))) float v8f;

// Monotone float -> u32 encoding so unsigned atomicMax == float max.
// enc(any finite float) > 0, so memset-0 is the identity element.
__device__ __forceinline__ unsigned fenc(float f) {
    unsigned u = __float_as_uint(f);
    return (u & 0x80000000u) ? ~u : (u | 0x80000000u);
}
__device__ __forceinline__ float fdec(unsigned u) {
    return __uint_as_float((u & 0x80000000u) ? (u & 0x7FFFFFFFu) : ~u);
}

// ---------------------------------------------------------------------------
// GEMM1: xl1 = x @ W1l, xr1 = x @ W1r   [50000x128] @ [128x64]
// One wave per 16-row M-tile; computes the full 16x64 strip for BOTH weight
// matrices using V_WMMA_F32_16X16X4_F32 (exact fp32). 32 K-steps of K=4.
// A 16x4 layout: lane half -> K pair; B 4x16: symmetric; C/D per ISA.
// ---------------------------------------------------------------------------
__global__ __launch_bounds__(32) void k_gemm1_wmma(
    const float* __restrict__ x, const float* __restrict__ Wl,
    const float* __restrict__ Wr, float* __restrict__ xl,
    float* __restrict__ xr) {
    const int mtile = blockIdx.x;          // 0..3124 (50000/16)
    const int lane  = threadIdx.x;
    const int half  = lane >> 4;           // 0/1
    const int mrow  = lane & 15;
    const int arow  = mtile * 16 + mrow;   // A row for this lane

    v8f accL[4] = {};
    v8f accR[4] = {};

    for (int k = 0; k < F_IN; k += 4) {
        const int kk = k + half * 2;
        v2f a;
        a.x = x[(size_t)arow * F_IN + kk + 0];
        a.y = x[(size_t)arow * F_IN + kk + 1];
#pragma unroll
        for (int t = 0; t < 4; ++t) {
            const int col = t * 16 + mrow;
            v2f bl, br;
            bl.x = Wl[(size_t)(kk + 0) * HC1 + col];
            bl.y = Wl[(size_t)(kk + 1) * HC1 + col];
            br.x = Wr[(size_t)(kk + 0) * HC1 + col];
            br.y = Wr[(size_t)(kk + 1) * HC1 + col];
            accL[t] = __builtin_amdgcn_wmma_f32_16x16x4_f32(
                false, a, false, bl, (short)0, accL[t], false, false);
            accR[t] = __builtin_amdgcn_wmma_f32_16x16x4_f32(
                false, a, false, br, (short)0, accR[t], false, false);
        }
    }

#pragma unroll
    for (int t = 0; t < 4; ++t) {
#pragma unroll
        for (int v = 0; v < 8; ++v) {
            const int row = mtile * 16 + v + 8 * half;
            const int col = t * 16 + mrow;
            xl[(size_t)row * HC1 + col] = accL[t][v];
            xr[(size_t)row * HC1 + col] = accR[t][v];
        }
    }
}

// ---------------------------------------------------------------------------
// Edge stage, layer 1. One thread per (edge, head).
// ---------------------------------------------------------------------------
__global__ __launch_bounds__(256) void k_edge1_logits(
    const int* __restrict__ ei, const float* __restrict__ xl1,
    const float* __restrict__ xr1, const float* __restrict__ att1,
    float* __restrict__ lg1, unsigned* __restrict__ m1) {
    const int t = blockIdx.x * 256 + threadIdx.x;
    if (t >= N_EDGES * H1) return;
    const int e = t >> 3, h = t & 7;
    const int src = ei[e], dst = ei[N_EDGES + e];
    const float4* pl = (const float4*)(xl1 + (size_t)src * HC1 + h * C1);
    const float4* pr = (const float4*)(xr1 + (size_t)dst * HC1 + h * C1);
    const float4 l0 = pl[0], l1 = pl[1], r0 = pr[0], r1 = pr[1];
    float s[8] = {l0.x + r0.x, l0.y + r0.y, l0.z + r0.z, l0.w + r0.w,
                  l1.x + r1.x, l1.y + r1.y, l1.z + r1.z, l1.w + r1.w};
    float logit = 0.f;
#pragma unroll
    for (int c = 0; c < 8; ++c) {
        const float v = s[c] > 0.f ? s[c] : NEG_SLOPE * s[c];
        logit += att1[h * C1 + c] * v;
    }
    lg1[t] = logit;
    atomicMax(m1 + (size_t)dst * H1 + h, fenc(logit));
}

__global__ __launch_bounds__(256) void k_edge1_agg(
    const int* __restrict__ ei, const float* __restrict__ xl1,
    const float* __restrict__ lg1, const unsigned* __restrict__ m1,
    float* __restrict__ den1, float* __restrict__ hacc) {
    const int t = blockIdx.x * 256 + threadIdx.x;
    if (t >= N_EDGES * H1) return;
    const int e = t >> 3, h = t & 7;
    const int src = ei[e], dst = ei[N_EDGES + e];
    const float p = expf(lg1[t] - fdec(m1[(size_t)dst * H1 + h]));
    atomicAdd(den1 + (size_t)dst * H1 + h, p);
    const float4* pl = (const float4*)(xl1 + (size_t)src * HC1 + h * C1);
    const float4 l0 = pl[0], l1 = pl[1];
    float* ob = hacc + (size_t)dst * HC1 + h * C1;
    atomicAdd(ob + 0, p * l0.x); atomicAdd(ob + 1, p * l0.y);
    atomicAdd(ob + 2, p * l0.z); atomicAdd(ob + 3, p * l0.w);
    atomicAdd(ob + 4, p * l1.x); atomicAdd(ob + 5, p * l1.y);
    atomicAdd(ob + 6, p * l1.z); atomicAdd(ob + 7, p * l1.w);
}

// h[n,j] = relu(acc/denom + bias); in place on hacc.
__global__ __launch_bounds__(256) void k_finalize1(
    const float* __restrict__ den1, const float* __restrict__ b1,
    float* __restrict__ hacc) {
    const int t = blockIdx.x * 256 + threadIdx.x;
    if (t >= N_NODES * HC1) return;
    const int n = t >> 6, j = t & 63;
    const float d = den1[(size_t)n * H1 + (j >> 3)];
    float v = (d > 0.f) ? hacc[t] / d : 0.f;
    v += b1[j];
    hacc[t] = v > 0.f ? v : 0.f;
}

// ---------------------------------------------------------------------------
// GEMM2: xl2 = h @ W2l, xr2 = h @ W2r   [50000x64] @ [64x10]  (VALU; tiny)
// ---------------------------------------------------------------------------
__global__ __launch_bounds__(256) void k_gemm2(
    const float* __restrict__ h, const float* __restrict__ Wl,
    const float* __restrict__ Wr, float* __restrict__ xl2,
    float* __restrict__ xr2) {
    const int n = blockIdx.x * 256 + threadIdx.x;
    if (n >= N_NODES) return;
    float sl[NC] = {}, sr[NC] = {};
    const float* hr = h + (size_t)n * HC1;
    for (int k = 0; k < HC1; ++k) {
        const float hv = hr[k];
#pragma unroll
        for (int c = 0; c < NC; ++c) {
            sl[c] += hv * Wl[k * NC + c];
            sr[c] += hv * Wr[k * NC + c];
        }
    }
#pragma unroll
    for (int c = 0; c < NC; ++c) {
        xl2[(size_t)n * NC + c] = sl[c];
        xr2[(size_t)n * NC + c] = sr[c];
    }
}

__global__ __launch_bounds__(256) void k_edge2_logits(
    const int* __restrict__ ei, const float* __restrict__ xl2,
    const float* __restrict__ xr2, const float* __restrict__ att2,
    float* __restrict__ lg2, unsigned* __restrict__ m2) {
    const int e = blockIdx.x * 256 + threadIdx.x;
    if (e >= N_EDGES) return;
    const int src = ei[e], dst = ei[N_EDGES + e];
    float logit = 0.f;
#pragma unroll
    for (int c = 0; c < NC; ++c) {
        float s = xl2[(size_t)src * NC + c] + xr2[(size_t)dst * NC + c];
        s = s > 0.f ? s : NEG_SLOPE * s;
        logit += att2[c] * s;
    }
    lg2[e] = logit;
    atomicMax(m2 + dst, fenc(logit));
}

__global__ __launch_bounds__(256) void k_edge2_agg(
    const int* __restrict__ ei, const float* __restrict__ xl2,
    const float* __restrict__ lg2, const unsigned* __restrict__ m2,
    float* __restrict__ den2, float* __restrict__ acc2) {
    const int e = blockIdx.x * 256 + threadIdx.x;
    if (e >= N_EDGES) return;
    const int src = ei[e], dst = ei[N_EDGES + e];
    const float p = expf(lg2[e] - fdec(m2[dst]));
    atomicAdd(den2 + dst, p);
#pragma unroll
    for (int c = 0; c < NC; ++c)
        atomicAdd(acc2 + (size_t)dst * NC + c, p * xl2[(size_t)src * NC + c]);
}

// out = log_softmax(acc/denom + bias)
__global__ __launch_bounds__(256) void k_finalize2(
    const float* __restrict__ acc2, const float* __restrict__ den2,
    const float* __restrict__ b2, float* __restrict__ out) {
    const int n = blockIdx.x * 256 + threadIdx.x;
    if (n >= N_NODES) return;
    const float d = den2[n];
    float v[NC];
    float mx = -3.0e38f;
#pragma unroll
    for (int c = 0; c < NC; ++c) {
        const float base = (d > 0.f) ? acc2[(size_t)n * NC + c] / d : 0.f;
        v[c] = base + b2[c];
        mx = v[c] > mx ? v[c] : mx;
    }
    float sum = 0.f;
#pragma unroll
    for (int c = 0; c < NC; ++c) sum += expf(v[c] - mx);
    const float lse = logf(sum);
#pragma unroll
    for (int c = 0; c < NC; ++c)
        out[(size_t)n * NC + c] = v[c] - mx - lse;
}

extern "C" void kernel_launch(void* const* d_in, const int* in_sizes, int n_in,
                              void* d_out, int out_size, void* d_ws, size_t ws_size,
                              hipStream_t stream) {
    (void)in_sizes; (void)n_in; (void)out_size; (void)ws_size;
    const float* x    = (const float*)d_in[0];
    const int*   ei   = (const int*)  d_in[1];
    const float* W1l  = (const float*)d_in[2];
    const float* W1r  = (const float*)d_in[3];
    const float* att1 = (const float*)d_in[4];
    const float* b1   = (const float*)d_in[5];
    const float* W2l  = (const float*)d_in[6];
    const float* W2r  = (const float*)d_in[7];
    const float* att2 = (const float*)d_in[8];
    const float* b2   = (const float*)d_in[9];
    float* out = (float*)d_out;

    // Workspace layout (floats). Zero-initialized region first (one memset).
    float* ws = (float*)d_ws;
    size_t off = 0;
    float*    hacc = ws + off; off += (size_t)N_NODES * HC1;   // zero
    unsigned* m1   = (unsigned*)(ws + off); off += (size_t)N_NODES * H1; // zero
    float*    den1 = ws + off; off += (size_t)N_NODES * H1;    // zero
    float*    acc2 = ws + off; off += (size_t)N_NODES * NC;    // zero
    unsigned* m2   = (unsigned*)(ws + off); off += N_NODES;    // zero
    float*    den2 = ws + off; off += N_NODES;                 // zero
    const size_t zero_floats = off;
    float* xl1 = ws + off; off += (size_t)N_NODES * HC1;
    float* xr1 = ws + off; off += (size_t)N_NODES * HC1;
    float* lg1 = ws + off; off += (size_t)N_EDGES * H1;
    float* xl2 = ws + off; off += (size_t)N_NODES * NC;
    float* xr2 = ws + off; off += (size_t)N_NODES * NC;
    float* lg2 = ws + off; off += (size_t)N_EDGES;

    hipMemsetAsync(ws, 0, zero_floats * sizeof(float), stream);

    // Layer 1
    k_gemm1_wmma<<<N_NODES / 16, 32, 0, stream>>>(x, W1l, W1r, xl1, xr1);
    {
        const int total = N_EDGES * H1;
        const int blocks = (total + 255) / 256;
        k_edge1_logits<<<blocks, 256, 0, stream>>>(ei, xl1, xr1, att1, lg1, m1);
        k_edge1_agg<<<blocks, 256, 0, stream>>>(ei, xl1, lg1, m1, den1, hacc);
    }
    k_finalize1<<<(N_NODES * HC1 + 255) / 256, 256, 0, stream>>>(den1, b1, hacc);

    // Layer 2
    k_gemm2<<<(N_NODES + 255) / 256, 256, 0, stream>>>(hacc, W2l, W2r, xl2, xr2);
    k_edge2_logits<<<(N_EDGES + 255) / 256, 256, 0, stream>>>(ei, xl2, xr2, att2, lg2, m2);
    k_edge2_agg<<<(N_EDGES + 255) / 256, 256, 0, stream>>>(ei, xl2, lg2, m2, den2, acc2);
    k_finalize2<<<(N_NODES + 255) / 256, 256, 0, stream>>>(acc2, den2, b2, out);
}